// ProxyFuserAttn_22840636080889
// MI455X (gfx1250) — compile-verified
//
#include <hip/hip_runtime.h>
#include <hip/hip_bf16.h>
#include <math.h>

typedef float v2f __attribute__((ext_vector_type(2)));
typedef float v4f __attribute__((ext_vector_type(4)));
typedef float v8f __attribute__((ext_vector_type(8)));
typedef unsigned int v4u __attribute__((ext_vector_type(4)));
typedef int v4i __attribute__((ext_vector_type(4)));
typedef int v8i __attribute__((ext_vector_type(8)));

// D = A(16x4) * B(4x16) + C, fp32 WMMA (CDNA5)
#define WMMA_F32(a, b, c) \
  __builtin_amdgcn_wmma_f32_16x16x4_f32(false, (a), false, (b), (short)0, (c), false, false)

static constexpr int CC = 512;

// ---------------------------------------------------------------------------
// TDM: async 2D tile load (global -> LDS), fp32 elements, LDS row padding of
// 4 dwords after every 64 dwords (tile_d0 must be 64 for the pad cadence).
// D# packing per CDNA5 ISA ch.8; 6-arg builtin form on this toolchain.
// ---------------------------------------------------------------------------
__device__ inline void tdm_load_2d_f32(const float* gsrc, unsigned int lds_addr,
                                       unsigned int tile_d0, unsigned int tile_d1,
                                       unsigned long long d0_stride_elems)
{
  const unsigned long long ga = (unsigned long long)(uintptr_t)gsrc;
  v4u g0;
  g0.x = 1u;                                   // count=1, user desc, no gather
  g0.y = lds_addr;                             // LDS byte address
  g0.z = (unsigned int)(ga & 0xffffffffu);     // global_addr[31:0]
  g0.w = (unsigned int)((ga >> 32) & 0x1ffffffu) | (2u << 30);  // [56:32] | type=2

  const unsigned int td0 = tile_d0;            // tensor dims == tile dims (in-bounds)
  const unsigned int td1 = tile_d1;
  unsigned int w0 = (2u << 16)                 // data_size: 4 bytes
                  | (1u << 20)                 // pad_enable
                  | (5u << 22)                 // pad_interval: every 64 dwords
                  | (3u << 25);                // pad_amount: 4 dwords
  unsigned int w1 = (td0 & 0xffffu) << 16;                       // tensor_dim0[15:0]
  unsigned int w2 = (td0 >> 16) | ((td1 & 0xffffu) << 16);       // dim0[31:16]|dim1[15:0]
  unsigned int w3 = (td1 >> 16) | (tile_d0 << 16);               // dim1[31:16]|tile_dim0
  unsigned int w4 = (tile_d1 & 0xffffu);                         // tile_dim1 (tile_dim2=0)
  unsigned int w5 = (unsigned int)(d0_stride_elems & 0xffffffffu);   // dim0_stride[31:0]
  unsigned int w6 = (unsigned int)((d0_stride_elems >> 32) & 0xffffu); // [47:32]
  unsigned int w7 = 0u;                                          // dim1_stride unused (2D)
  v8i g1; g1[0]=(int)w0; g1[1]=(int)w1; g1[2]=(int)w2; g1[3]=(int)w3;
          g1[4]=(int)w4; g1[5]=(int)w5; g1[6]=(int)w6; g1[7]=(int)w7;
  v4i gz4 = {0, 0, 0, 0};
  v8i gz8 = {0, 0, 0, 0, 0, 0, 0, 0};
  __builtin_amdgcn_tensor_load_to_lds(g0, g1, gz4, gz4, gz8, 0);
}

// ---------------------------------------------------------------------------
// BN (eval) + ragged->dense pad : tok[b*1024+l][c], float4 vectorized
// ---------------------------------------------------------------------------
__global__ __launch_bounds__(128)
void k_bn_pad(const float* __restrict__ x, const int* __restrict__ lens,
              const float* __restrict__ g, const float* __restrict__ be,
              const float* __restrict__ mu, const float* __restrict__ va,
              float* __restrict__ tok)
{
  const int row = blockIdx.x;          // 0..8191
  const int b = row >> 10, l = row & 1023;
  int off = 0;
  for (int i = 0; i < b; ++i) off += lens[i];
  const bool valid = l < lens[b];
  const int c = threadIdx.x * 4;       // 128 threads x float4 = 512 channels
  v4f o = {0.f, 0.f, 0.f, 0.f};
  if (valid) {
    const v4f xv = *(const v4f*)(x + (size_t)(off + l) * CC + c);
    const v4f m4 = *(const v4f*)(mu + c);
    const v4f v4 = *(const v4f*)(va + c);
    const v4f g4 = *(const v4f*)(g + c);
    const v4f b4 = *(const v4f*)(be + c);
#pragma unroll
    for (int j = 0; j < 4; ++j)
      o[j] = (xv[j] - m4[j]) * rsqrtf(v4[j] + 1e-5f) * g4[j] + b4[j];
  }
  *(v4f*)(tok + (size_t)row * CC + c) = o;
}

// ---------------------------------------------------------------------------
// fp32 WMMA GEMM: D[m][n] = sum_k A[m][k] * W[n][k] + bias[n], epilogue fused.
// Block: 256 threads = 8 waves in 4(M) x 2(N); block tile 64x128.
// W tile [128][64] staged in LDS by the Tensor Data Mover, double-buffered,
// LDS rows padded to 68 floats (TDM pad) for bank-conflict-free B fragments.
// ---------------------------------------------------------------------------
enum { EPI_QKV = 0, EPI_OUTPROJ = 1, EPI_GELU = 2, EPI_RES = 3 };

template <int EPI>
__global__ __launch_bounds__(256)
void k_gemm(const float* __restrict__ A, int lda,
            const float* __restrict__ W, int ldw,
            const float* __restrict__ bias,
            float* __restrict__ o0, float* __restrict__ o1, float* __restrict__ o2,
            const float* __restrict__ aux, const int* __restrict__ lens,
            const float* __restrict__ bng, const float* __restrict__ bnb,
            const float* __restrict__ bnm, const float* __restrict__ bnv,
            int M, int N, int K)
{
  __shared__ float Wt[2][128 * 68];      // 2 x 34.8 KB, TDM-padded rows

  const int tid  = threadIdx.x;
  const int lane = tid & 31, wave = tid >> 5;
  const int l16  = lane & 15, lh  = lane >> 4;
  const int wm   = wave & 3,  wn  = wave >> 2;
  const int nbn  = N >> 7;
  const int bm   = blockIdx.x / nbn, bn = blockIdx.x % nbn;
  const int m0   = bm * 64 + wm * 16;
  const int n0   = bn * 128 + wn * 64;

  const float* Arow  = A + (size_t)(m0 + l16) * lda;
  const float* Wbase = W + (size_t)(bn * 128) * ldw;
  const int nkc = K >> 6;                // 64-wide K chunks

  if (wave == 0) {
    tdm_load_2d_f32(Wbase, (unsigned int)(uintptr_t)&Wt[0][0], 64, 128,
                    (unsigned long long)ldw);
    __builtin_amdgcn_s_wait_tensorcnt(0);
  }
  __syncthreads();

  v8f acc[4] = {};
  int cur = 0;
  for (int kc = 0; kc < nkc; ++kc) {
    if (wave == 0 && kc + 1 < nkc)       // prefetch next chunk via TDM
      tdm_load_2d_f32(Wbase + (kc + 1) * 64,
                      (unsigned int)(uintptr_t)&Wt[cur ^ 1][0], 64, 128,
                      (unsigned long long)ldw);

    const float* Wl = &Wt[cur][0];
    const int kbase = kc * 64;
    __builtin_prefetch(Arow + kbase + 64, 0, 0);
#pragma unroll
    for (int ku = 0; ku < 64; ku += 16) {          // 4 WMMA K-steps per group
      v2f af[4], bf[4][4];
#pragma unroll
      for (int u = 0; u < 4; ++u) {                // batched loads first
        const int kl = ku + u * 4 + 2 * lh;        // k within chunk
        af[u] = *(const v2f*)(Arow + kbase + kl);
#pragma unroll
        for (int t = 0; t < 4; ++t)
          bf[u][t] = *(const v2f*)(Wl + (size_t)(wn * 64 + t * 16 + l16) * 68 + kl);
      }
#pragma unroll
      for (int u = 0; u < 4; ++u) {                // then 16 back-to-back WMMAs
        acc[0] = WMMA_F32(af[u], bf[u][0], acc[0]);
        acc[1] = WMMA_F32(af[u], bf[u][1], acc[1]);
        acc[2] = WMMA_F32(af[u], bf[u][2], acc[2]);
        acc[3] = WMMA_F32(af[u], bf[u][3], acc[3]);
      }
    }
    if (wave == 0) __builtin_amdgcn_s_wait_tensorcnt(0);
    __syncthreads();
    cur ^= 1;
  }

  // per-thread constants for ragged mapping (16-row tile never crosses a batch)
  int offb = 0, lenb = 0;
  if (EPI == EPI_OUTPROJ) {
    const int bb = m0 >> 10;
    for (int i = 0; i < bb; ++i) offb += lens[i];
    lenb = lens[bb];
  }

#pragma unroll
  for (int t = 0; t < 4; ++t) {
#pragma unroll
    for (int r = 0; r < 8; ++r) {
      const int row = m0 + r + 8 * lh;      // D layout: VGPR r -> rows r / r+8
      const int col = n0 + t * 16 + l16;
      float v = acc[t][r] + bias[col];
      if (EPI == EPI_QKV) {
        const int which = col >> 9;          // 0=q 1=k 2=v
        const int c = col & 511;
        const int h = c >> 6, d = c & 63;
        const int b8 = row >> 10, ll = row & 1023;
        const size_t idx = ((size_t)(b8 * 8 + h) * 1024 + ll) * 64 + d;
        if (which == 0)      o0[idx] = v * 0.125f;   // fold 1/sqrt(dh) into q
        else if (which == 1) o1[idx] = v;
        else                 o2[idx] = v;
      } else if (EPI == EPI_OUTPROJ) {
        const int ll = row & 1023;
        if (ll < lenb) {
          const size_t idx = (size_t)(offb + ll) * 512 + col;
          const float outv = v + aux[idx];   // + residual(features), unpadded
          o0[idx] = outv;
          // fused FFN pre-norm: xn = BN(outv)
          o1[idx] = (outv - bnm[col]) * rsqrtf(bnv[col] + 1e-5f) * bng[col] + bnb[col];
        }
      } else if (EPI == EPI_GELU) {
        o0[(size_t)row * N + col] = 0.5f * v * (1.f + erff(v * 0.70710678118654752f));
      } else {  // EPI_RES
        const size_t idx = (size_t)row * 512 + col;
        o0[idx] = v + aux[idx];
      }
    }
  }
}

// ---------------------------------------------------------------------------
// Fused attention: one block = 16 q-rows of one batch, loops all 8 heads.
// Score row-block + head-mean accumulator in LDS (~137 KB of 320 KB/WGP).
// rpe read exactly once; [B,H,L,L] never touches HBM.
// ---------------------------------------------------------------------------
__global__ __launch_bounds__(256)
void k_attn(const float* __restrict__ q, const float* __restrict__ kmat,
            const float* __restrict__ vmat, const float* __restrict__ rpe,
            const int* __restrict__ lens,
            float* __restrict__ ctx, float* __restrict__ attnw)
{
  __shared__ float S[16][1028];     // scores/probs for current head (bank-safe pad)
  __shared__ float ACCS[16][1028];  // sum_h attn/8
  __shared__ float CTXP[2][16][68]; // ctx partials (K split across wave halves)

  const int tid  = threadIdx.x;
  const int lane = tid & 31, wave = tid >> 5;
  const int l16  = lane & 15, lh  = lane >> 4;
  const int b    = blockIdx.x >> 6;
  const int qt   = blockIdx.x & 63;
  const int lenb = lens[b];

  for (int i = tid; i < 16 * 1028; i += 256) (&ACCS[0][0])[i] = 0.f;
  __syncthreads();

  for (int h = 0; h < 8; ++h) {
    const int bh = b * 8 + h;
    const float* qb = q    + (size_t)bh * 1024 * 64;
    const float* kb = kmat + (size_t)bh * 1024 * 64;
    const float* vb = vmat + (size_t)bh * 1024 * 64;
    const float* rp = rpe  + (size_t)bh * 1024 * 1024 + (size_t)(qt * 16 + 8 * lh) * 1024;

    // q fragments for all 16 K-steps (A 16x4 layout: float2, half-wave K split)
    v2f af[16];
#pragma unroll
    for (int ks = 0; ks < 16; ++ks)
      af[ks] = *(const v2f*)(qb + (size_t)(qt * 16 + l16) * 64 + ks * 4 + 2 * lh);

    // --- scores: each wave computes 8 key tiles of 16; batch loads, then WMMAs
    for (int i = 0; i < 8; ++i) {
      const int n0 = (wave * 8 + i) * 16;
      v2f bf[16];
#pragma unroll
      for (int ks = 0; ks < 16; ++ks)
        bf[ks] = *(const v2f*)(kb + (size_t)(n0 + l16) * 64 + ks * 4 + 2 * lh);
      v8f acc = {};
#pragma unroll
      for (int ks = 0; ks < 16; ++ks)
        acc = WMMA_F32(af[ks], bf[ks], acc);

      const int col = n0 + l16;
      const float mb = (col < lenb) ? 0.f : -1e9f;   // additive key mask
      float rv[8];
#pragma unroll
      for (int r = 0; r < 8; ++r)                    // batched rpe loads (4KB stride)
        rv[r] = rp[(size_t)r * 1024 + col];
#pragma unroll
      for (int r = 0; r < 8; ++r)
        S[r + 8 * lh][col] = acc[r] + rv[r] + mb;
    }
    __syncthreads();

    // --- softmax (2 rows per wave) + accumulate head-mean ---
    for (int rr = 0; rr < 2; ++rr) {
      const int row = wave * 2 + rr;
      float m = -3.4e38f;
      for (int c = lane; c < 1024; c += 32) m = fmaxf(m, S[row][c]);
#pragma unroll
      for (int s = 16; s; s >>= 1) m = fmaxf(m, __shfl_xor(m, s, 32));
      float sum = 0.f;
      for (int c = lane; c < 1024; c += 32) {
        const float e = __expf(S[row][c] - m);
        S[row][c] = e;
        sum += e;
      }
#pragma unroll
      for (int s = 16; s; s >>= 1) sum += __shfl_xor(sum, s, 32);
      const float inv = 1.f / sum;
      for (int c = lane; c < 1024; c += 32) {
        const float p = S[row][c] * inv;
        S[row][c] = p;
        ACCS[row][c] += 0.125f * p;     // mean over 8 heads
      }
    }
    __syncthreads();

    // --- ctx = P @ V : waves 0-3 ntile with K[0,512), waves 4-7 K[512,1024) ---
    {
      const int nt = wave & 3, kh = wave >> 2;
      const int n0 = nt * 16;
      const int kbeg = kh * 512;
      v8f acc = {};
      for (int k = kbeg; k < kbeg + 512; k += 32) {   // 8 steps: loads then WMMAs
        v2f a8[8], b8[8];
#pragma unroll
        for (int u = 0; u < 8; ++u) {
          const int kk = k + u * 4 + 2 * lh;
          a8[u] = *(const v2f*)(&S[l16][kk]);
          b8[u].x = vb[(size_t)kk * 64 + n0 + l16];
          b8[u].y = vb[(size_t)(kk + 1) * 64 + n0 + l16];
        }
#pragma unroll
        for (int u = 0; u < 8; ++u) acc = WMMA_F32(a8[u], b8[u], acc);
      }
#pragma unroll
      for (int r = 0; r < 8; ++r)
        CTXP[kh][r + 8 * lh][n0 + l16] = acc[r];
    }
    __syncthreads();

    // combine partials, write ctx in [B,L,C] concat layout
    for (int i = tid; i < 16 * 64; i += 256) {
      const int row = i >> 6, col = i & 63;
      ctx[(size_t)(b * 1024 + qt * 16 + row) * 512 + h * 64 + col] =
          CTXP[0][row][col] + CTXP[1][row][col];
    }
    __syncthreads();
  }

  // write the head-averaged attention block (written exactly once)
  for (int i = tid; i < 16 * 1024; i += 256) {
    const int row = i >> 10, col = i & 1023;
    attnw[(size_t)(b * 1024 + qt * 16 + row) * 1024 + col] = ACCS[row][col];
  }
}

// ---------------------------------------------------------------------------
extern "C" void kernel_launch(void* const* d_in, const int* in_sizes, int n_in,
                              void* d_out, int out_size, void* d_ws, size_t ws_size,
                              hipStream_t stream)
{
  (void)in_sizes; (void)n_in; (void)out_size; (void)ws_size;
  const float* features = (const float*)d_in[0];
  const int*   lens     = (const int*)d_in[2];
  const float* rpe      = (const float*)d_in[3];
  const float* ng = (const float*)d_in[4];
  const float* nb = (const float*)d_in[5];
  const float* nm = (const float*)d_in[6];
  const float* nv = (const float*)d_in[7];
  const float* wqkv = (const float*)d_in[8];
  const float* bqkv = (const float*)d_in[9];
  const float* wo = (const float*)d_in[10];
  const float* bo = (const float*)d_in[11];
  const float* fg = (const float*)d_in[12];
  const float* fb = (const float*)d_in[13];
  const float* fm = (const float*)d_in[14];
  const float* fv = (const float*)d_in[15];
  const float* w1 = (const float*)d_in[16];
  const float* b1 = (const float*)d_in[17];
  const float* w2 = (const float*)d_in[18];
  const float* b2 = (const float*)d_in[19];

  float* ws   = (float*)d_ws;
  float* tok  = ws;                              // 8192*512
  float* qb   = tok  + (size_t)8192 * 512;       // 64*1024*64
  float* kb   = qb   + (size_t)64 * 1024 * 64;
  float* vb   = kb   + (size_t)64 * 1024 * 64;
  float* ctx  = vb   + (size_t)64 * 1024 * 64;   // 8192*512
  float* aout = ctx  + (size_t)8192 * 512;       // 6144*512
  float* xn   = aout + (size_t)6144 * 512;       // 6144*512
  float* hbuf = xn   + (size_t)6144 * 512;       // 6144*1024

  float* out_tok  = (float*)d_out;                    // [6144,512]
  float* out_attn = out_tok + (size_t)6144 * 512;     // [8,1024,1024]

  k_bn_pad<<<8192, 128, 0, stream>>>(features, lens, ng, nb, nm, nv, tok);
  k_gemm<EPI_QKV><<<(8192 / 64) * (1536 / 128), 256, 0, stream>>>(
      tok, 512, wqkv, 512, bqkv, qb, kb, vb, nullptr, lens,
      nullptr, nullptr, nullptr, nullptr, 8192, 1536, 512);
  k_attn<<<512, 256, 0, stream>>>(qb, kb, vb, rpe, lens, ctx, out_attn);
  k_gemm<EPI_OUTPROJ><<<(8192 / 64) * (512 / 128), 256, 0, stream>>>(
      ctx, 512, wo, 512, bo, aout, xn, nullptr, features, lens,
      fg, fb, fm, fv, 8192, 512, 512);
  k_gemm<EPI_GELU><<<(6144 / 64) * (1024 / 128), 256, 0, stream>>>(
      xn, 512, w1, 512, b1, hbuf, nullptr, nullptr, nullptr, nullptr,
      nullptr, nullptr, nullptr, nullptr, 6144, 1024, 512);
  k_gemm<EPI_RES><<<(6144 / 64) * (512 / 128), 256, 0, stream>>>(
      hbuf, 1024, w2, 1024, b2, out_tok, nullptr, nullptr, aout, nullptr,
      nullptr, nullptr, nullptr, nullptr, 6144, 512, 1024);
}